// LocalGlance_25486335934728
// MI455X (gfx1250) — compile-verified
//
#include <hip/hip_runtime.h>
#include <stdint.h>

// ---------------------------------------------------------------------------
// SSIM (3x3 uniform window, stride 3, pad 1) on 8x1x2048x2048 fp32 -> scalar.
// Memory-bound: 256 MB single-pass stream -> ~11 us floor @ 23.3 TB/s.
// Data path: global_load_async_to_lds_b128 staging + s_wait_asynccnt (CDNA5),
// conflict-free stride-3 LDS reads, deterministic two-pass reduction.
// ---------------------------------------------------------------------------

#define BATCH 8
#define IH    2048
#define IW    2048
#define OH    683
#define OW    683
#define OXB   3        // column blocks per output row (256 + 256 + 171)
#define TPB   256
#define LDSW  768      // bulk input columns staged per block (3 * 256)

#define C1F   (0.0001f)  // 0.01^2
#define C2F   (0.0009f)  // 0.03^2
#define INV9  (0.111111111111111111f)

// Only enforce the builtin's presence in the DEVICE pass; the host pass
// parses device code too but doesn't declare amdgcn builtins.
#if defined(__HIP_DEVICE_COMPILE__)
#if !__has_builtin(__builtin_amdgcn_global_load_async_to_lds_b128)
#error "device toolchain lacks __builtin_amdgcn_global_load_async_to_lds_b128"
#endif
#endif

typedef int v4i __attribute__((ext_vector_type(4)));

__device__ __forceinline__ void async_cp16(const float* g, float* l) {
#if __has_builtin(__builtin_amdgcn_global_load_async_to_lds_b128)
  // 16B global -> LDS async copy, tracked by ASYNCcnt. (device pass)
  __builtin_amdgcn_global_load_async_to_lds_b128((v4i*)g, (v4i*)l, 0, 0);
#else
  // host-pass parse-only fallback (never code-generated for gfx1250)
  *(v4i*)l = *(const v4i*)g;
#endif
}

__device__ __forceinline__ void wait_async0() {
#if __has_builtin(__builtin_amdgcn_s_wait_asynccnt)
  __builtin_amdgcn_s_wait_asynccnt(0);
#else
  asm volatile("s_wait_asynccnt 0" ::: "memory");
#endif
}

__global__ __launch_bounds__(TPB) void ssim_tile_kernel(
    const float* __restrict__ img1, const float* __restrict__ img2,
    float* __restrict__ partial) {
  __shared__ float lds[2][3][LDSW];
  __shared__ float red[TPB];

  const int tid = threadIdx.x;
  int blk = (int)blockIdx.x;
  const int oxb = blk % OXB;  blk /= OXB;
  const int oy  = blk % OH;
  const int b   = blk / OH;

  const int ox0 = oxb * TPB;
  const int c0  = 3 * ox0;                       // first bulk input column (mult of 4)
  const int ncols = min(LDSW, IW - c0);          // 768 or 512 (last block)
  const int nc4 = ncols >> 2;                    // float4 count per row
  const int iy0 = 3 * oy - 1;                    // top input row (may be -1 -> pad)

  const size_t ibase = (size_t)b * IH * IW;
  const float* i1 = img1 + ibase;
  const float* i2 = img2 + ibase;

  // ---- async bulk stage: 2 images x 3 rows x ncols floats, all 16B aligned ----
  const int total = 2 * 3 * nc4;
  for (int t = tid; t < total; t += TPB) {
    const int im  = t / (3 * nc4);
    const int rem = t - im * (3 * nc4);
    const int r   = rem / nc4;
    const int c4  = rem - r * nc4;
    int iy = iy0 + r;
    if (iy < 0) iy = 0;                          // safe row; compute skips pad row
    const float* src = (im == 0 ? i1 : i2) + (size_t)iy * IW + c0 + 4 * c4;
    async_cp16(src, &lds[im][r][4 * c4]);
  }
  wait_async0();
  __syncthreads();

  // ---- per-thread output pixel ----
  const int ox = ox0 + tid;
  float ssim = 0.0f;
  if (ox < OW) {
    float s1 = 0.f, s2 = 0.f, s11 = 0.f, s22 = 0.f, s12 = 0.f;
    const int jb = 3 * tid;
#pragma unroll
    for (int r = 0; r < 3; ++r) {
      if (iy0 + r < 0) continue;                 // zero-pad top row (uniform per block)
#pragma unroll
      for (int d = -1; d <= 1; ++d) {
        const int j = jb + d;
        float x1, x2;
        if (j < 0) {                             // only thread 0's left tap
          if (c0 == 0) { x1 = 0.f; x2 = 0.f; }   // zero-pad left column
          else {
            const size_t off = (size_t)(iy0 + r) * IW + (c0 - 1);
            x1 = i1[off]; x2 = i2[off];
          }
        } else {                                 // stride-3 LDS reads: gcd(3,64)=1 -> no conflicts
          x1 = lds[0][r][j];
          x2 = lds[1][r][j];
        }
        s1 += x1; s2 += x2;
        s11 = fmaf(x1, x1, s11);
        s22 = fmaf(x2, x2, s22);
        s12 = fmaf(x1, x2, s12);
      }
    }
    const float mu1 = s1 * INV9, mu2 = s2 * INV9;
    const float mu1s = mu1 * mu1, mu2s = mu2 * mu2, mu12 = mu1 * mu2;
    const float sg1  = s11 * INV9 - mu1s;
    const float sg2  = s22 * INV9 - mu2s;
    const float sg12 = s12 * INV9 - mu12;
    const float num = (2.f * mu12 + C1F) * (2.f * sg12 + C2F);
    const float den = (mu1s + mu2s + C1F) * (sg1 + sg2 + C2F);
    ssim = num / den;
  }

  // ---- deterministic block reduction ----
  red[tid] = ssim;
  __syncthreads();
#pragma unroll
  for (int s = TPB / 2; s > 0; s >>= 1) {
    if (tid < s) red[tid] += red[tid + s];
    __syncthreads();
  }
  if (tid == 0) partial[blockIdx.x] = red[0];
}

__global__ __launch_bounds__(1024) void ssim_final_kernel(
    const float* __restrict__ partial, int n, float* __restrict__ out) {
  __shared__ double red[1024];
  double acc = 0.0;
  for (int i = threadIdx.x; i < n; i += 1024) acc += (double)partial[i];
  red[threadIdx.x] = acc;
  __syncthreads();
#pragma unroll
  for (int s = 512; s > 0; s >>= 1) {
    if (threadIdx.x < s) red[threadIdx.x] += red[threadIdx.x + s];
    __syncthreads();
  }
  if (threadIdx.x == 0)
    out[0] = (float)(red[0] / ((double)BATCH * OH * OW));
}

extern "C" void kernel_launch(void* const* d_in, const int* in_sizes, int n_in,
                              void* d_out, int out_size, void* d_ws, size_t ws_size,
                              hipStream_t stream) {
  const float* img1 = (const float*)d_in[0];
  const float* img2 = (const float*)d_in[1];
  // d_in[2] (window) is uniform 1/9 by construction; folded into INV9.
  float* partial = (float*)d_ws;                 // 16392 floats (64 KB) of scratch
  const int nblocks = BATCH * OH * OXB;          // 8 * 683 * 3 = 16392
  ssim_tile_kernel<<<nblocks, TPB, 0, stream>>>(img1, img2, partial);
  ssim_final_kernel<<<1, 1024, 0, stream>>>(partial, nblocks, (float*)d_out);
}